// RepulsionLoss_20409684590746
// MI455X (gfx1250) — compile-verified
//
#include <hip/hip_runtime.h>

typedef __attribute__((ext_vector_type(2))) float v2f;
typedef __attribute__((ext_vector_type(8))) float v8f;

#define NPTS      8192
#define KNN       10          // neighbors kept (reference K)
#define RADIUS_F  0.07f
#define EPS_F     1e-12f
#define ROWSTRIDE 20          // LDS row stride in words: 80B rows (16B aligned, no ds bank conflicts)
#define BIGF      3.4e38f

// One wave handles 16 rows of one batch; streams all 512 column tiles.
__launch_bounds__(32)
__global__ void RepulsionLoss_kernel(const float* __restrict__ pts,
                                     float* __restrict__ out,
                                     float scale) {
    const int wg      = blockIdx.x;
    const int batch   = wg >> 9;        // 512 row-tiles per batch (8192/16)
    const int rowTile = wg & 511;
    const int lane    = threadIdx.x;    // 0..31 (wave32)
    const int n16     = lane & 15;
    const bool hi     = lane >= 16;

    const float* __restrict__ base = pts + (size_t)batch * NPTS * 3;

    // ---- A fragment (16x4 f32, MxK): A[m] = (-2x, -2y, -2z, |p_m|^2) ----
    // Layout (ISA 7.12.2): lanes 0-15 hold K=0 (v0) / K=1 (v1); lanes 16-31 hold K=2 / K=3.
    const int r0 = rowTile * 16;
    const float* __restrict__ rp = base + (size_t)(r0 + n16) * 3;
    const float px = rp[0], py = rp[1], pz = rp[2];
    const float sqx = px * px + py * py + pz * pz;
    v2f afrag;
    afrag.x = hi ? (-2.0f * pz) : (-2.0f * px);
    afrag.y = hi ? sqx          : (-2.0f * py);

    __shared__ __align__(16) float tile[16 * ROWSTRIDE];
    __shared__ float mergebuf[16 * 22];

    // Register-resident sorted top-(K+1) of SQUARED distances (ascending).
    float best[KNN + 1];
#pragma unroll
    for (int i = 0; i <= KNN; ++i) best[i] = BIGF;

    const int myRow  = lane >> 1;  // 2 threads per row
    const int myHalf = lane & 1;   // each owns 8 of the 16 tile columns
    const int rowBase = hi ? 8 : 0;

    for (int ct = 0; ct < NPTS / 16; ++ct) {
        // ---- B fragment (4x16 f32, KxN): column n = (y0, y1, y2, 1) ----
        const float* __restrict__ qp = base + (size_t)(ct * 16 + n16) * 3;
        const float qx = qp[0], qy = qp[1], qz = qp[2];
        const float sqy = qx * qx + qy * qy + qz * qz;
        v2f bfrag;
        bfrag.x = hi ? qz   : qx;   // rows K=2 / K=0
        bfrag.y = hi ? 1.0f : qy;   // rows K=3 / K=1

        // C preloaded with |q_n|^2 (N = lane%16 in C/D layout) -> D = |p|^2+|q|^2-2 p.q
        v8f acc;
#pragma unroll
        for (int i = 0; i < 8; ++i) acc[i] = sqy;

        acc = __builtin_amdgcn_wmma_f32_16x16x4_f32(
            /*neg_a=*/false, afrag, /*neg_b=*/false, bfrag,
            /*c_mod=*/(short)0, acc, /*reuse_a=*/false, /*reuse_b=*/false);

        // Scatter D tile to LDS: VGPR r -> row (hi ? r+8 : r), col = lane%16
#pragma unroll
        for (int r = 0; r < 8; ++r)
            tile[(rowBase + r) * ROWSTRIDE + n16] = acc[r];
        __syncthreads();

        // Each thread pulls its 8 contiguous candidates (2x ds_load_b128)
        const float4* tp = (const float4*)&tile[myRow * ROWSTRIDE + myHalf * 8];
        const float4 c0 = tp[0];
        const float4 c1 = tp[1];
        const float cand[8] = {c0.x, c0.y, c0.z, c0.w, c1.x, c1.y, c1.z, c1.w};

#pragma unroll
        for (int j = 0; j < 8; ++j) {
            const float d2 = cand[j];
            if (d2 < best[KNN]) {          // common path: 1 compare, branch out
                float v = d2;
#pragma unroll
                for (int i = 0; i <= KNN; ++i) {   // sorted insert (min/max chain)
                    const float lo = fminf(best[i], v);
                    const float hv = fmaxf(best[i], v);
                    best[i] = lo;
                    v = hv;
                }
            }
        }
        __syncthreads();   // protect LDS tile before next iteration's stores
    }

    // ---- Merge the two half-row top-11 lists, drop self (rank 0), reduce ----
#pragma unroll
    for (int i = 0; i <= KNN; ++i)
        mergebuf[myRow * 22 + myHalf * 11 + i] = best[i];
    __syncthreads();

    if (lane < 16) {
        const float* la = &mergebuf[lane * 22];
        const float* lb = la + 11;
        int i = 0, j = 0;
        float sum = 0.0f;
        for (int k = 0; k <= KNN; ++k) {           // k+1 = 11 smallest of 22
            const float va = (i < 11) ? la[i] : BIGF;
            const float vb = (j < 11) ? lb[j] : BIGF;
            float v;
            if (va <= vb) { v = va; ++i; } else { v = vb; ++j; }
            if (k > 0) {                            // rank 0 == self distance
                const float d = sqrtf(fmaxf(v, EPS_F));
                sum += fmaxf(RADIUS_F - d, 0.0f);
            }
        }
        atomicAdd(out, sum * scale);                // global_atomic_add_f32
    }
}

__global__ void RepulsionLoss_zero(float* __restrict__ out) {
    out[0] = 0.0f;
}

extern "C" void kernel_launch(void* const* d_in, const int* in_sizes, int n_in,
                              void* d_out, int out_size, void* d_ws, size_t ws_size,
                              hipStream_t stream) {
    (void)n_in; (void)out_size; (void)d_ws; (void)ws_size;
    const float* pts = (const float*)d_in[0];
    float* out = (float*)d_out;

    const int B = in_sizes[0] / (NPTS * 3);                 // 4
    const float scale = 1.0f / ((float)B * (float)NPTS * (float)KNN);

    RepulsionLoss_zero<<<1, 1, 0, stream>>>(out);
    RepulsionLoss_kernel<<<B * (NPTS / 16), 32, 0, stream>>>(pts, out, scale);
}